// LRCComputer_12369505812590
// MI455X (gfx1250) — compile-verified
//
#include <hip/hip_runtime.h>

// ---------------------------------------------------------------------------
// AEV (ANI symmetry functions) for gfx1250 / MI455X.
//   radial:  out[s*16+f]   = sum_j onehot(sp_j==s) * 0.25*exp(-16(d-ShfR_f)^2)*fc(d,Rcr)
//   angular: out[64+b*32+f] = sum_pairs onehot(bin==b) * 2*((1+cos(th-ShfZ))/2)^32
//                              * exp(-8(dd-ShfA)^2) * fc(d1,Rca)*fc(d2,Rca)
// Both one-hot contractions run as V_WMMA_F32_16X16X4_F32 GEMMs (f32 in /
// f32 accumulate == reference precision path). One wave owns one atom.
// Per-j / per-pair scalars are computed once (wave-parallel) and staged in
// LDS so the WMMA fragment-build loops do only feature-specific math.
// ---------------------------------------------------------------------------

typedef float v2f __attribute__((ext_vector_type(2)));
typedef float v8f __attribute__((ext_vector_type(8)));

#define C_MOL   32
#define N_ATOM  64
#define NSPEC   4
#define KMAX    32
#define RAD_SUB 16
#define RAD_LEN 64
#define ANG_SUB 32
#define NBINS   10
#define AEV_LEN 384
#define RCR     5.2f
#define RCA     3.5f
#define PI_F    3.14159265358979f

__device__ __forceinline__ float fcut(float d, float invRc) {
  // caller guarantees d <= Rc (or masks the result)
  return 0.5f * __cosf(PI_F * d * invRc) + 0.5f;
}

__global__ __launch_bounds__(256)
void aev_kernel(const int* __restrict__ gSpecies,
                const float* __restrict__ gCoords,
                float* __restrict__ gOut) {
  // molecule coordinates / species staged once per block
  __shared__ float sPX[N_ATOM], sPY[N_ATOM], sPZ[N_ATOM];
  __shared__ int   sSp[N_ATOM];
  // per-wave (= per-atom) compacted neighbor records (d <= Rca)
  __shared__ float nbX[8][KMAX], nbY[8][KMAX], nbZ[8][KMAX];
  __shared__ float nbD[8][KMAX], nbF[8][KMAX];
  __shared__ int   nbS[8][KMAX];
  // per-wave full distance row + premasked 0.25*fc(d,Rcr) for the radial GEMM
  __shared__ float dAll[8][N_ATOM];
  __shared__ float fcAll[8][N_ATOM];
  // per-wave staged pair scalars for the angular GEMM (32-pair sweeps)
  __shared__ float pCT[8][KMAX], pST[8][KMAX], pDD[8][KMAX], pFCJ[8][KMAX];
  __shared__ int   pBIN[8][KMAX];

  const int tid  = threadIdx.x;
  const int lane = tid & 31;
  const int wave = tid >> 5;
  const int mol  = blockIdx.x >> 3;   // 32 molecules
  const int grp  = blockIdx.x & 7;    // 8 atom-groups per molecule
  const int atom = grp * 8 + wave;    // one wave per atom

  // ---- stage molecule into LDS -------------------------------------------
  for (int a = tid; a < N_ATOM; a += blockDim.x) {
    const float* p = gCoords + (size_t)(mol * N_ATOM + a) * 3;
    sPX[a] = p[0]; sPY[a] = p[1]; sPZ[a] = p[2];
    sSp[a] = gSpecies[mol * N_ATOM + a];
  }
  __syncthreads();

  const float xi = sPX[atom], yi = sPY[atom], zi = sPZ[atom];

  // ---- phase 1: distances, radial fc, neighbor compaction ----------------
  int cnt = 0;
  #pragma unroll
  for (int half = 0; half < 2; ++half) {
    int j = half * 32 + lane;
    float dx = xi - sPX[j], dy = yi - sPY[j], dz = zi - sPZ[j];
    float d  = sqrtf(dx*dx + dy*dy + dz*dz);
    bool self = (j == atom);
    dAll[wave][j]  = self ? 1e9f : d;           // poison diagonal
    // premasked 0.25*fc(d,Rcr): carries the (~eye & d<=Rcr) mask exactly,
    // and 0.25*fc is an exact scaling so the final product rounds as in ref
    fcAll[wave][j] = (!self && d <= RCR) ? 0.25f * fcut(d, 1.0f / RCR) : 0.f;
    bool isNb = (!self) && (d <= RCA);
    unsigned int bal = (unsigned int)__ballot(isNb);
    int pos = cnt + __popc(bal & ((1u << lane) - 1u));
    if (isNb && pos < KMAX) {
      nbX[wave][pos] = dx; nbY[wave][pos] = dy; nbZ[wave][pos] = dz;
      nbD[wave][pos] = d;
      nbF[wave][pos] = fcut(d, 1.0f / RCA);
      nbS[wave][pos] = sSp[j];
    }
    cnt += __popc(bal);
  }
  const int M = (cnt < KMAX) ? cnt : KMAX;
  __syncthreads();   // uniform barrier: all LDS records visible

  // Fragment geometry shared by both GEMMs (f32 WMMA 16x16x4 layout):
  //   lane<16 : A holds (M=lane,   K=kBase..kBase+1) with kBase=0
  //   lane>=16: A holds (M=lane-16,K=kBase..kBase+1) with kBase=2
  //   B mirrors: v0 = rows {0,2}, v1 = rows {1,3}; N = lane&15
  const int hiHalf = lane >> 4;
  const int kBase  = hiHalf * 2;
  const int nCol   = lane & 15;

  float* outAtom = gOut + (size_t)(mol * N_ATOM + atom) * AEV_LEN;

  // ---- phase 2: radial AEV via WMMA  (M=16 feat, K=64 atoms, N=4 species)-
  {
    const float shfr = 0.9f + 0.26875f * (float)(lane & 15);  // ShfR[f]
    v8f acc = {0.f, 0.f, 0.f, 0.f, 0.f, 0.f, 0.f, 0.f};
    for (int c4 = 0; c4 < 16; ++c4) {        // 16 k-chunks of 4 atoms
      int j0 = c4 * 4 + kBase, j1 = j0 + 1;
      float t0 = dAll[wave][j0] - shfr, t1 = dAll[wave][j1] - shfr;
      v2f A; A.x = fcAll[wave][j0] * __expf(-16.f * t0 * t0);
             A.y = fcAll[wave][j1] * __expf(-16.f * t1 * t1);
      v2f B; B.x = (sSp[j0] == nCol) ? 1.f : 0.f;
             B.y = (sSp[j1] == nCol) ? 1.f : 0.f;
      acc = __builtin_amdgcn_wmma_f32_16x16x4_f32(false, A, false, B,
                                                  (short)0, acc, false, false);
    }
    if (nCol < NSPEC) {
      #pragma unroll
      for (int r = 0; r < 8; ++r) {
        int f = r + hiHalf * 8;              // D: vgpr r = rows r / r+8
        outAtom[nCol * RAD_SUB + f] = acc[r];
      }
    }
  }

  // ---- phase 3: angular AEV via WMMA (two 16-feat tiles, K=pairs, N=10) --
  {
    const int P = M * (M - 1) / 2;           // valid pairs (<= 496)
    // per-lane feature constants: f = a*8 + z, z = lane&7, a = tile base
    const float shfz = (float)(2 * (lane & 7) + 1) * (PI_F / 16.f);
    const float cz = __cosf(shfz), sz = __sinf(shfz);
    const int   a0 = (lane >> 3) & 1;
    const float sa0 = 0.9f + 0.65f * (float)a0;        // tile0: ShfA[a0]
    const float sa1 = 0.9f + 0.65f * (float)(a0 + 2);  // tile1: ShfA[a0+2]

    v8f acc0 = {0.f,0.f,0.f,0.f,0.f,0.f,0.f,0.f};
    v8f acc1 = {0.f,0.f,0.f,0.f,0.f,0.f,0.f,0.f};

    for (int base = 0; base < P; base += KMAX) {
      // -- pre-pass: lane p computes pair (base+p)'s scalars once ----------
      {
        int p   = base + lane;
        bool vp = (p < P);
        int pc  = vp ? p : 0;                // pad reads pair 0 (valid: P>0)
        // decode pc -> (jj < kk): k-major enumeration so the first P entries
        // of the K=32 triangle are exactly the pairs over the M neighbors
        int kk = (int)((1.0f + sqrtf(8.0f * (float)pc + 1.0f)) * 0.5f);
        int jj = pc - (kk * (kk - 1)) / 2;
        if (jj < 0)        { kk -= 1; jj = pc - (kk * (kk - 1)) / 2; }
        else if (jj >= kk) { kk += 1; jj = pc - (kk * (kk - 1)) / 2; }

        float x1 = nbX[wave][jj], y1 = nbY[wave][jj], z1 = nbZ[wave][jj];
        float x2 = nbX[wave][kk], y2 = nbY[wave][kk], z2 = nbZ[wave][kk];
        float d1 = nbD[wave][jj], d2 = nbD[wave][kk];
        float ct = 0.95f * (x1*x2 + y1*y2 + z1*z2) / (d1 * d2);  // cos(theta)
        float st = sqrtf(fmaxf(0.f, 1.f - ct * ct));             // sin(theta)
        int s1 = nbS[wave][jj], s2 = nbS[wave][kk];
        int mn = (s1 < s2) ? s1 : s2, mx = (s1 < s2) ? s2 : s1;
        pCT[wave][lane]  = ct;
        pST[wave][lane]  = st;
        pDD[wave][lane]  = 0.5f * (d1 + d2);
        pFCJ[wave][lane] = vp ? 2.f * nbF[wave][jj] * nbF[wave][kk] : 0.f;
        pBIN[wave][lane] = mn * NSPEC - (mn * (mn - 1)) / 2 + (mx - mn);
      }
      __builtin_amdgcn_wave_barrier();  // LDS is in-order per wave; fence
                                        // compiler scheduling across phases
      // -- WMMA sweep over (up to) 32 staged pairs = 8 k-chunks ------------
      int limit = P - base; if (limit > KMAX) limit = KMAX;
      int nCh = (limit + 3) >> 2;
      for (int c4 = 0; c4 < nCh; ++c4) {
        float av[4], bv[2];
        #pragma unroll
        for (int q = 0; q < 2; ++q) {
          int lp  = c4 * 4 + kBase + q;      // local staged-pair slot
          float ct = pCT[wave][lp], st = pST[wave][lp];
          float dd = pDD[wave][lp], fcj2 = pFCJ[wave][lp];  // 0 for pads
          int  bin = pBIN[wave][lp];
          // ((1 + cos(theta - ShfZ))/2)^32 via 5 squarings (no arccos)
          float b0 = 0.5f * (1.f + ct * cz + st * sz);
          float f1 = b0 * b0; f1 *= f1; f1 *= f1; f1 *= f1; f1 *= f1;
          float ta = dd - sa0, tb = dd - sa1;
          float common = fcj2 * f1;
          av[q]     = common * __expf(-8.f * ta * ta);  // tile0 feature
          av[2 + q] = common * __expf(-8.f * tb * tb);  // tile1 feature
          bv[q]     = (lp < limit && bin == nCol) ? 1.f : 0.f;
        }
        v2f A0; A0.x = av[0]; A0.y = av[1];
        v2f A1; A1.x = av[2]; A1.y = av[3];
        v2f B;  B.x  = bv[0]; B.y  = bv[1];
        acc0 = __builtin_amdgcn_wmma_f32_16x16x4_f32(false, A0, false, B,
                                                     (short)0, acc0, false, false);
        acc1 = __builtin_amdgcn_wmma_f32_16x16x4_f32(false, A1, false, B,
                                                     (short)0, acc1, false, false);
      }
      __builtin_amdgcn_wave_barrier();  // keep next sweep's stores behind reads
    }

    if (nCol < NBINS) {
      #pragma unroll
      for (int r = 0; r < 8; ++r) {
        int fl = r + hiHalf * 8;
        outAtom[RAD_LEN + nCol * ANG_SUB + fl]      = acc0[r];  // f = 0..15
        outAtom[RAD_LEN + nCol * ANG_SUB + 16 + fl] = acc1[r];  // f = 16..31
      }
    }
  }
}

extern "C" void kernel_launch(void* const* d_in, const int* in_sizes, int n_in,
                              void* d_out, int out_size, void* d_ws, size_t ws_size,
                              hipStream_t stream) {
  (void)in_sizes; (void)n_in; (void)out_size; (void)d_ws; (void)ws_size;
  const int*   species = (const int*)d_in[0];
  const float* coords  = (const float*)d_in[1];
  float*       out     = (float*)d_out;
  // 256 blocks x 256 threads: 8 waves/block, one wave per atom (32*64 atoms)
  aev_kernel<<<dim3(C_MOL * (N_ATOM / 8)), dim3(256), 0, stream>>>(species, coords, out);
}